// TokenCorrector_5935644803459
// MI455X (gfx1250) — compile-verified
//
#include <hip/hip_runtime.h>
#include <stdint.h>

// ---------------------------------------------------------------------------
// TokenCorrector for MI455X (gfx1250, wave32)
//
// out[b,n,:] = image_token[b,n,:] + cnt(b,n) * delta[b,:]
//   delta[b] = strength * is_rare[b] * (l2norm(text_cls[b]) - l2norm(pooled[b]))
//   cnt(b,n) = #occurrences of n in topk_idx[b,:]   (duplicates accumulate)
//
// Bandwidth-bound: ~302MB traffic -> ~13us floor at 23.3 TB/s.
// CDNA5 paths used:
//   * v_wmma_f32_16x16x4_f32 for the row-norm Gram diagonals (tiny, free)
//   * global_load_async_to_lds_b128 / global_store_async_from_lds_b128 +
//     s_wait_asynccnt for the bulk row copy (no VGPR round trip)
// ---------------------------------------------------------------------------

typedef __attribute__((ext_vector_type(2))) float v2f;
typedef __attribute__((ext_vector_type(8))) float v8f;
typedef __attribute__((ext_vector_type(4))) int   v4i;

// Probe-derived async builtin signature:
//   (v4i* generic/global side, __shared__ v4i* LDS side, imm int, imm int)
#define LPTR(p) ((__attribute__((address_space(3))) v4i*)(p))

#if __has_builtin(__builtin_amdgcn_global_load_async_to_lds_b128) && \
    __has_builtin(__builtin_amdgcn_global_store_async_from_lds_b128)
#define HAVE_ASYNC_LDS 1
#else
#define HAVE_ASYNC_LDS 0
#warning "async global<->LDS builtins unavailable on this toolchain; VGPR copy fallback in use"
#endif

__device__ __forceinline__ void wait_asynccnt0() {
#if __has_builtin(__builtin_amdgcn_s_wait_asynccnt)
  __builtin_amdgcn_s_wait_asynccnt(0);
#else
  asm volatile("s_wait_asynccnt 0" ::: "memory");
#endif
}

// ---------------------------------------------------------------------------
// Kernel A: delta[b,:] for a tile of 16 batch rows per 32-lane wave.
// Norms come from the diagonal of X * X^T accumulated with
// V_WMMA_F32_16X16X4_F32. For the documented 16x4 f32 A layout
// (lane L holds row L%16, K = 2*(L>=16) + vgpr) and the mirrored 4x16 B
// layout, operand registers for A and for B==A^T are IDENTICAL, so we feed
// the same v2f to both sources. Diagonal (M==N) of the 16x16 f32 C tile:
//   row m (m<8)  -> C[m]   at lane m
//   row m (m>=8) -> C[m-8] at lane m+16
// ---------------------------------------------------------------------------
__global__ __launch_bounds__(32) void tc_delta_kernel(
    const float* __restrict__ text_cls,   // [B,D]
    const float* __restrict__ pooled,     // [B,D]
    const int*   __restrict__ is_rare,    // [B]
    const float* __restrict__ strength,   // [1]
    float*       __restrict__ delta,      // [B,D] (workspace)
    int B, int D) {
  const int lane = threadIdx.x;
  const int m0   = blockIdx.x * 16;
  const int r    = lane & 15;
  const int kap  = (lane >> 4) << 1;                 // 0 for lanes 0-15, 2 for 16-31
  const int rowc = min(m0 + r, B - 1);               // clamp (B==64 -> exact)

  float dt = 0.0f, dp = 0.0f;                        // per-lane diagonal element

#if __has_builtin(__builtin_amdgcn_wmma_f32_16x16x4_f32)
#define DIAG_SRC(m) (((m) < 8) ? (m) : ((m) + 16))
  const float* trow = text_cls + (size_t)rowc * D + kap;
  const float* prow = pooled   + (size_t)rowc * D + kap;
  v8f ct = {};
  v8f cp = {};
  for (int k0 = 0; k0 < D; k0 += 4) {
    v2f ta = *(const v2f*)(trow + k0);
    v2f pa = *(const v2f*)(prow + k0);
    // C += A * A^T ; same registers serve A-layout and B-layout of X^T.
    ct = __builtin_amdgcn_wmma_f32_16x16x4_f32(false, ta, false, ta,
                                               (short)0, ct, false, false);
    cp = __builtin_amdgcn_wmma_f32_16x16x4_f32(false, pa, false, pa,
                                               (short)0, cp, false, false);
  }
  {
    const int j = (lane < 8) ? lane : ((lane >= 24) ? (lane - 24) : 0);
    dt = (j == 0) ? ct[0] : (j == 1) ? ct[1] : (j == 2) ? ct[2] : (j == 3) ? ct[3]
       : (j == 4) ? ct[4] : (j == 5) ? ct[5] : (j == 6) ? ct[6] : ct[7];
    dp = (j == 0) ? cp[0] : (j == 1) ? cp[1] : (j == 2) ? cp[2] : (j == 3) ? cp[3]
       : (j == 4) ? cp[4] : (j == 5) ? cp[5] : (j == 6) ? cp[6] : cp[7];
  }
#else
#define DIAG_SRC(m) (m)
#warning "wmma_f32_16x16x4_f32 builtin unavailable; scalar norm fallback"
  if (lane < 16) {
    const float* tr = text_cls + (size_t)rowc * D;
    const float* pr = pooled   + (size_t)rowc * D;
    for (int d2 = 0; d2 < D; ++d2) {
      dt += tr[d2] * tr[d2];
      dp += pr[d2] * pr[d2];
    }
  }
#endif

  const float s = strength[0];
  for (int m = 0; m < 16; ++m) {
    const int b = m0 + m;
    if (b >= B) break;
    float nt = sqrtf(__shfl(dt, DIAG_SRC(m), 32));
    float np = sqrtf(__shfl(dp, DIAG_SRC(m), 32));
    nt = fmaxf(nt, 1e-12f);                          // torch F.normalize eps
    np = fmaxf(np, 1e-12f);
    const float sc  = s * ((is_rare[b] != 0) ? 1.0f : 0.0f);
    const float rnt = sc / nt;
    const float rnp = sc / np;
    const float* tr = text_cls + (size_t)b * D;
    const float* pr = pooled   + (size_t)b * D;
    float*       dr = delta    + (size_t)b * D;
    for (int d = lane; d < D; d += 32)
      dr[d] = tr[d] * rnt - pr[d] * rnp;
  }
}

// ---------------------------------------------------------------------------
// Kernel B: one block per (b, n) token row of D floats.
//   cnt == 0 (vast majority): async stream memory -> LDS -> memory, 128b/lane
//   cnt  > 0: VGPR path, out = in + cnt * delta[b]   (exact duplicate handling)
// The branch is block-uniform so EXEC stays all-ones.
// ---------------------------------------------------------------------------
__global__ __launch_bounds__(256) void tc_copy_scatter_kernel(
    const float* __restrict__ img,     // [B,N,D]
    const float* __restrict__ delta,   // [B,D]
    const int*   __restrict__ idx,     // [B,K]
    float*       __restrict__ out,     // [B,N,D]
    int N, int D, int K) {
  const int n = blockIdx.x;
  const int b = blockIdx.y;

  int cnt = 0;
  const int* row_idx = idx + (size_t)b * K;
  for (int k = 0; k < K; ++k) cnt += (row_idx[k] == n) ? 1 : 0;

  const size_t off       = ((size_t)b * N + n) * (size_t)D;
  const int    t         = threadIdx.x;
  const int    vecPerRow = D >> 2;                   // float4 per row

  __shared__ __align__(16) char buf[256 * 16];
  (void)buf;

  if (cnt == 0) {
#if HAVE_ASYNC_LDS
    if (vecPerRow == 256) {                          // D == 1024: one b128/lane
      const float4* src = (const float4*)(img + off);
      float4*       dst = (float4*)(out + off);
      __builtin_amdgcn_global_load_async_to_lds_b128(
          (v4i*)(src + t), LPTR(buf + (size_t)t * 16), 0, 0);
      wait_asynccnt0();                              // LDS write visible
      __builtin_amdgcn_global_store_async_from_lds_b128(
          (v4i*)(dst + t), LPTR(buf + (size_t)t * 16), 0, 0);
      // s_endpgm implicitly waits all counters; no trailing wait needed.
      return;
    }
#endif
    for (int e = t; e < vecPerRow; e += 256)
      ((float4*)(out + off))[e] = ((const float4*)(img + off))[e];
  } else {
    const float   c  = (float)cnt;
    const float4* dl = (const float4*)(delta + (size_t)b * D);
    for (int e = t; e < vecPerRow; e += 256) {
      float4 v  = ((const float4*)(img + off))[e];
      float4 dv = dl[e];
      v.x += c * dv.x;
      v.y += c * dv.y;
      v.z += c * dv.z;
      v.w += c * dv.w;
      ((float4*)(out + off))[e] = v;
    }
  }
}

// ---------------------------------------------------------------------------
// Host launcher. Input order (setup_inputs dict):
//   0: image_token f32 [B,N,D]   1: text_cls f32 [B,D]   2: topk_idx int [B,K]
//   3: selected_pooled f32 [B,D] 4: is_rare int [B]      5: strength f32 [1]
// ---------------------------------------------------------------------------
extern "C" void kernel_launch(void* const* d_in, const int* in_sizes, int n_in,
                              void* d_out, int out_size, void* d_ws, size_t ws_size,
                              hipStream_t stream) {
  const float* image    = (const float*)d_in[0];
  const float* text     = (const float*)d_in[1];
  const int*   topk     = (const int*)d_in[2];
  const float* pooled   = (const float*)d_in[3];
  const int*   rare     = (const int*)d_in[4];
  const float* strength = (const float*)d_in[5];
  float*       out      = (float*)d_out;

  const int B = in_sizes[4];                 // 64
  const int D = in_sizes[1] / B;             // 1024
  const int N = in_sizes[0] / (B * D);       // 576
  const int K = in_sizes[2] / B;             // 16

  float* delta = (float*)d_ws;               // B*D floats = 256 KB scratch

  tc_delta_kernel<<<dim3((B + 15) / 16), dim3(32), 0, stream>>>(
      text, pooled, rare, strength, delta, B, D);

  tc_copy_scatter_kernel<<<dim3(N, B), dim3(256), 0, stream>>>(
      image, delta, topk, out, N, D, K);
}